// QuantizedSoftmax2_13563506721249
// MI455X (gfx1250) — compile-verified
//
#include <hip/hip_runtime.h>
#include <cstdint>

// Quantized softmax over last axis (N=1024) of an (8,8,1024,1024) f32 tensor.
// Memory-bound: 512 MB min traffic => ~22 us at 23.3 TB/s. One wave32 per row,
// rows staged into LDS once via CDNA5 async global->LDS b128 copies
// (double-buffered, ASYNCcnt-tracked), 3 LDS passes (max / sum / emit),
// nontemporal b128 stores for the write-once output.
//
// All exp2() calls operate on exact small integers, so they are replaced by
// full-rate integer exponent construction ((k+127)<<23 bit pattern) instead of
// quarter-rate v_exp_f32 trans ops; the final dequant (*2^-eb) is folded into
// the exponent as well.

typedef float v4f __attribute__((ext_vector_type(4)));
typedef int v4i __attribute__((ext_vector_type(4)));

#define WAVES_PER_BLOCK 8
#define ROW_N 1024
#define CHUNKS 8        // 8 chunks * 32 lanes * 4 floats = 1024 elements
#define V4_PER_ROW 256  // ROW_N / 4

#if __has_builtin(__builtin_amdgcn_global_load_async_to_lds_b128)
#define HAVE_ASYNC 1
#else
#define HAVE_ASYNC 0
#endif

#if HAVE_ASYNC
// Inline asm (not builtin) so the "memory" clobber pins LDS reads after the wait.
#define WAIT_ASYNC(n) asm volatile("s_wait_asynccnt %0" ::"i"(n) : "memory")
#else
#define WAIT_ASYNC(n) ((void)0)
#endif

// Stage one 4KB row into LDS. Per lane: base = row + lane*16B; the 24-bit
// instruction offset (c*512B) applies to BOTH the global and LDS address
// (CDNA5 ISA GV-mode pseudocode), so one VGPR base pair serves
// all 8 chunks (lowers to an s_clause of 8 global_load_async_to_lds_b128).
__device__ __forceinline__ void stage_row(const float* __restrict__ gsrc,
                                          v4f* ldst) {
#if HAVE_ASYNC
  auto g = (__attribute__((address_space(1))) v4i*)(uintptr_t)gsrc;
  auto l = (__attribute__((address_space(3))) v4i*)(uintptr_t)ldst;
  __builtin_amdgcn_global_load_async_to_lds_b128(g, l, 0, 0);
  __builtin_amdgcn_global_load_async_to_lds_b128(g, l, 512, 0);
  __builtin_amdgcn_global_load_async_to_lds_b128(g, l, 1024, 0);
  __builtin_amdgcn_global_load_async_to_lds_b128(g, l, 1536, 0);
  __builtin_amdgcn_global_load_async_to_lds_b128(g, l, 2048, 0);
  __builtin_amdgcn_global_load_async_to_lds_b128(g, l, 2560, 0);
  __builtin_amdgcn_global_load_async_to_lds_b128(g, l, 3072, 0);
  __builtin_amdgcn_global_load_async_to_lds_b128(g, l, 3584, 0);
#else
  const v4f* gv = (const v4f*)gsrc;
#pragma unroll
  for (int c = 0; c < CHUNKS; ++c) ldst[c * 32] = gv[c * 32];
#endif
}

// exp2 of an exact-integer float t (valid when 0 <= t <= 127), 0 when t < 0.
__device__ __forceinline__ float pow2_or_zero(float t) {
  const int i = (int)t;  // exact: t is integer-valued
  const float v = __int_as_float((i + 127) << 23);
  return (i >= 0) ? v : 0.0f;
}

// floor(2^u) * 2^-eb for integer-valued u (u <= eb here), 0 when u < 0.
// Valid because u >= 0 => result exponent in [127-eb, 127]: no denormals.
__device__ __forceinline__ float pow2_dequant_or_zero(float u, int koff) {
  const int i = (int)u;  // exact: u is integer-valued
  const float v = __int_as_float((i + koff) << 23);  // koff = 127 - eb
  return (i >= 0) ? v : 0.0f;
}

__global__ __launch_bounds__(WAVES_PER_BLOCK * 32) void
qsoftmax_rows_kernel(const float* __restrict__ x, const int* __restrict__ ebp,
                     float* __restrict__ out, int nrows) {
  __shared__ v4f lds[WAVES_PER_BLOCK][2][V4_PER_ROW];  // 64 KB
  const int lane = (int)(threadIdx.x & 31u);
  const int wave = (int)(threadIdx.x >> 5u);

  const int ebi = *ebp;            // 8
  const float eb = (float)ebi;
  const int koff = 127 - ebi;      // folds *2^-eb into the output exponent

  const int step = (int)gridDim.x * WAVES_PER_BLOCK;
  int row = (int)blockIdx.x * WAVES_PER_BLOCK + wave;
  int cur = 0;

  if (row < nrows)
    stage_row(x + (size_t)row * ROW_N + (size_t)lane * 4, &lds[wave][cur][lane]);

  for (; row < nrows; row += step) {
    const int nrow = row + step;
    if (nrow < nrows) {
      // Prefetch next row into the other buffer, then wait until only those
      // 8 async ops remain outstanding => current buffer is complete
      // (async loads complete in order).
      stage_row(x + (size_t)nrow * ROW_N + (size_t)lane * 4,
                &lds[wave][cur ^ 1][lane]);
      WAIT_ASYNC(8);
    } else {
      WAIT_ASYNC(0);
    }

    const v4f* buf = &lds[wave][cur][0];

    // ---- pass 1: row max (lane-local, then wave32 xor-shuffle tree) ----
    float m = -3.402823466e38f;
#pragma unroll
    for (int c = 0; c < CHUNKS; ++c) {
      v4f v = buf[c * 32 + lane];
      m = fmaxf(m, fmaxf(fmaxf(v.x, v.y), fmaxf(v.z, v.w)));
    }
#pragma unroll
    for (int off = 16; off >= 1; off >>= 1)
      m = fmaxf(m, __shfl_xor(m, off, 32));

    // ---- pass 2: sum of 2^t over t = x - max + 2*eb >= 0 (exact ints) ----
    const float bias = 2.0f * eb - m;  // t = x + bias, integer-valued, <= 2*eb
    float s = 0.0f;
#pragma unroll
    for (int c = 0; c < CHUNKS; ++c) {
      v4f v = buf[c * 32 + lane];
      s += pow2_or_zero(v.x + bias);
      s += pow2_or_zero(v.y + bias);
      s += pow2_or_zero(v.z + bias);
      s += pow2_or_zero(v.w + bias);
    }
#pragma unroll
    for (int off = 16; off >= 1; off >>= 1)
      s += __shfl_xor(s, off, 32);

    // shift = round-to-even(log2(sum)) - eb; sum >= 2^(2*eb) => shift >= eb,
    // so masked lanes (t < 0) have u = t - shift < -eb < 0 and emit 0:
    // the emit pass is branch-free.
    const float shift = rintf(log2f(s)) - eb;
    const float sb = bias - shift;

    // ---- pass 3: emit floor(2^(t-shift)) * 2^-eb via exponent math ----
    float* orow = out + (size_t)row * ROW_N;
#pragma unroll
    for (int c = 0; c < CHUNKS; ++c) {
      v4f v = buf[c * 32 + lane];
      v4f o;
      o.x = pow2_dequant_or_zero(v.x + sb, koff);
      o.y = pow2_dequant_or_zero(v.y + sb, koff);
      o.z = pow2_dequant_or_zero(v.z + sb, koff);
      o.w = pow2_dequant_or_zero(v.w + sb, koff);
      __builtin_nontemporal_store(o, (v4f*)(orow + c * 128 + lane * 4));
    }

    cur ^= 1;
  }
}

extern "C" void kernel_launch(void* const* d_in, const int* in_sizes, int n_in,
                              void* d_out, int out_size, void* d_ws,
                              size_t ws_size, hipStream_t stream) {
  const float* x = (const float*)d_in[0];
  const int* eb = (const int*)d_in[1];
  float* out = (float*)d_out;

  const int nrows = in_sizes[0] / ROW_N;  // 65536 for the reference shape
  int blocks = (nrows + WAVES_PER_BLOCK - 1) / WAVES_PER_BLOCK;
  if (blocks > 2048) blocks = 2048;  // ~4 rows/wave => prefetch pipeline active
  if (blocks < 1) blocks = 1;

  qsoftmax_rows_kernel<<<dim3(blocks), dim3(WAVES_PER_BLOCK * 32), 0, stream>>>(
      x, eb, out, nrows);
}